// SlotAttentionV2_81595788689923
// MI455X (gfx1250) — compile-verified
//
#include <hip/hip_runtime.h>
#include <cstdint>

#define B_ 32
#define L_ 4096
#define C_ 256
#define V_ 5
#define T_ 3
#define TL_ 256
#define EPS_ 1e-8f
#define LN_EPS_ 1e-5f

typedef float v2f __attribute__((ext_vector_type(2)));
typedef float v4f __attribute__((ext_vector_type(4)));
typedef float v8f __attribute__((ext_vector_type(8)));

// ---------------------------------------------------------------------------
// Kernel 1: fused LayerNorm + q/v projections (WMMA f32 16x16x4) + mean partials
// grid (L/64, B), block 256 (8 waves). Block tile: 64 rows x 256 cols.
// Wave (4 x 2 layout): 16 rows x 128 cols for BOTH q and v => 16 x v8f accum.
// ---------------------------------------------------------------------------
__global__ __launch_bounds__(256)
void proj_qv_kernel(const float* __restrict__ feat,
                    const float* __restrict__ Wq,
                    const float* __restrict__ Wv,
                    const float* __restrict__ g_in,
                    const float* __restrict__ b_in,
                    float* __restrict__ qout,
                    float* __restrict__ vout,
                    float* __restrict__ base_part)
{
  __shared__ float xs[64][260];   // pad: A-frag column reads conflict-free
  __shared__ float wq[32][272];   // pad: (272*2)%64==32 -> K and K+2 rows hit disjoint banks
  __shared__ float wv[32][272];

  const int b    = blockIdx.y;
  const int lt   = blockIdx.x;
  const int l0   = lt * 64;
  const int tid  = threadIdx.x;
  const int lane = tid & 31;
  const int wave = tid >> 5;

  // ---- load 64x256 feature tile (fully contiguous 64KB) ----
  const float* fptr = feat + ((size_t)b * L_ + l0) * C_;
  for (int i = tid; i < 64 * 64; i += 256) {
    int r = i >> 6, c4 = i & 63;
    v4f t = ((const v4f*)fptr)[(size_t)r * 64 + c4];
    xs[r][c4 * 4 + 0] = t.x; xs[r][c4 * 4 + 1] = t.y;
    xs[r][c4 * 4 + 2] = t.z; xs[r][c4 * 4 + 3] = t.w;
  }
  __syncthreads();

  // ---- raw-feature column partial sums (for slot init mean over L) ----
  {
    float s = 0.f;
    #pragma unroll 8
    for (int r = 0; r < 64; ++r) s += xs[r][tid];
    base_part[((size_t)b * 64 + lt) * C_ + tid] = s;
  }

  // ---- LayerNorm: 8 rows per wave, wave32 butterfly reduction ----
  for (int r8 = 0; r8 < 8; ++r8) {
    int r = wave * 8 + r8;
    float s = 0.f, s2 = 0.f;
    #pragma unroll
    for (int c = lane; c < C_; c += 32) { float x = xs[r][c]; s += x; s2 += x * x; }
    #pragma unroll
    for (int o = 16; o > 0; o >>= 1) {
      s  += __shfl_xor(s,  o, 32);
      s2 += __shfl_xor(s2, o, 32);
    }
    float m    = s * (1.f / C_);
    float varv = s2 * (1.f / C_) - m * m;
    float inv  = rsqrtf(varv + LN_EPS_);
    #pragma unroll
    for (int c = lane; c < C_; c += 32)
      xs[r][c] = (xs[r][c] - m) * inv * g_in[c] + b_in[c];
  }

  // ---- WMMA GEMM ----
  const int waveM = wave & 3;
  const int waveN = wave >> 1 >> 1;          // wave/4
  const int rowb  = waveM * 16;
  const int colb  = waveN * 128;
  const int l16   = lane & 15;
  const int khi   = (lane >= 16) ? 2 : 0;    // A/B VGPR layout: K0/K1 lanes 0-15, K2/K3 lanes 16-31
  const int mofs  = (lane >= 16) ? 8 : 0;    // D layout: VGPR r holds rows r (lo lanes) and r+8 (hi lanes)

  v8f accq[8] = {};
  v8f accv[8] = {};

  for (int kc = 0; kc < C_; kc += 32) {
    __syncthreads();
    // stage 32 K-rows of Wq and Wv (coalesced b128 loads)
    for (int i = tid; i < 32 * 64; i += 256) {
      int r = i >> 6, c4 = i & 63;
      ((v4f*)&wq[r][0])[c4] = ((const v4f*)(Wq + (size_t)(kc + r) * C_))[c4];
      ((v4f*)&wv[r][0])[c4] = ((const v4f*)(Wv + (size_t)(kc + r) * C_))[c4];
    }
    if (kc + 32 < C_) {  // global_prefetch_b8 next chunk (32KB each, 128B/lane)
      __builtin_prefetch(Wq + (size_t)(kc + 32) * C_ + tid * 32, 0, 0);
      __builtin_prefetch(Wv + (size_t)(kc + 32) * C_ + tid * 32, 0, 0);
    }
    __syncthreads();

    #pragma unroll
    for (int kk = 0; kk < 32; kk += 4) {
      v2f a;
      a.x = xs[rowb + l16][kc + kk + khi];
      a.y = xs[rowb + l16][kc + kk + khi + 1];
      #pragma unroll
      for (int nt = 0; nt < 8; ++nt) {
        int col = colb + nt * 16 + l16;
        v2f bq, bv;
        bq.x = wq[kk + khi][col];     bq.y = wq[kk + khi + 1][col];
        bv.x = wv[kk + khi][col];     bv.y = wv[kk + khi + 1][col];
        accq[nt] = __builtin_amdgcn_wmma_f32_16x16x4_f32(
            false, a, false, bq, (short)0, accq[nt], false, false);
        accv[nt] = __builtin_amdgcn_wmma_f32_16x16x4_f32(
            false, a, false, bv, (short)0, accv[nt], false, false);
      }
    }
  }

  // ---- store q/v tiles ----
  const size_t obase = ((size_t)b * L_ + l0) * C_;
  #pragma unroll
  for (int nt = 0; nt < 8; ++nt) {
    int col = colb + nt * 16 + l16;
    #pragma unroll
    for (int r = 0; r < 8; ++r) {
      size_t idx = obase + (size_t)(rowb + r + mofs) * C_ + col;
      qout[idx] = accq[nt][r];
      vout[idx] = accv[nt][r];
    }
  }
}

// ---------------------------------------------------------------------------
// Kernel 2: slots = mean_L(features) + noise * |noise_scale|
// ---------------------------------------------------------------------------
__global__ __launch_bounds__(256)
void slots_init_kernel(const float* __restrict__ base_part,
                       const float* __restrict__ noise,
                       const float* __restrict__ noise_scale,
                       float* __restrict__ slots)
{
  const int b = blockIdx.x;
  const int c = threadIdx.x;
  float s = 0.f;
  for (int p = 0; p < 64; ++p) s += base_part[((size_t)b * 64 + p) * C_ + c];
  const float base = s * (1.f / L_);
  const float ns   = fabsf(noise_scale[0]);
  for (int v = 0; v < V_; ++v)
    slots[((size_t)b * V_ + v) * C_ + c] =
        base + noise[((size_t)b * V_ + v) * C_ + c] * ns;
}

// ---------------------------------------------------------------------------
// deterministic block reduction (sum, sumsq) over 256 threads
// ---------------------------------------------------------------------------
__device__ __forceinline__ void block_reduce_2(float& s, float& s2, float* red)
{
  const int lane = threadIdx.x & 31;
  const int wave = threadIdx.x >> 5;
  #pragma unroll
  for (int o = 16; o > 0; o >>= 1) {
    s  += __shfl_xor(s,  o, 32);
    s2 += __shfl_xor(s2, o, 32);
  }
  if (lane == 0) { red[wave] = s; red[8 + wave] = s2; }
  __syncthreads();
  float a = 0.f, a2 = 0.f;
  #pragma unroll
  for (int i = 0; i < 8; ++i) { a += red[i]; a2 += red[8 + i]; }
  __syncthreads();
  s = a; s2 = a2;
}

// ---------------------------------------------------------------------------
// Kernel 3: k = LN(slots) @ Wk   (grid B, 256 threads; Wk stays in L2)
// ---------------------------------------------------------------------------
__global__ __launch_bounds__(256)
void compute_k_kernel(const float* __restrict__ slots,
                      const float* __restrict__ Wk,
                      const float* __restrict__ g_s,
                      const float* __restrict__ b_s,
                      float* __restrict__ kout)
{
  __shared__ float sn[V_][C_];
  __shared__ float red[16];
  const int b   = blockIdx.x;
  const int tid = threadIdx.x;
  for (int s = 0; s < V_; ++s) {
    float x  = slots[((size_t)b * V_ + s) * C_ + tid];
    float ss = x, s2 = x * x;
    block_reduce_2(ss, s2, red);
    float m   = ss * (1.f / C_);
    float inv = rsqrtf(s2 * (1.f / C_) - m * m + LN_EPS_);
    sn[s][tid] = (x - m) * inv * g_s[tid] + b_s[tid];
  }
  __syncthreads();
  for (int s = 0; s < V_; ++s) {
    float acc = 0.f;
    for (int j = 0; j < C_; ++j) acc += sn[s][j] * Wk[(size_t)j * C_ + tid];
    kout[((size_t)b * V_ + s) * C_ + tid] = acc;
  }
}

// ---------------------------------------------------------------------------
// Kernel 4: attention streaming pass (bandwidth-bound; reads q,v once each)
// grid (L/TL, B), block 256. Per-wave update partials -> ws.
// ---------------------------------------------------------------------------
__global__ __launch_bounds__(256)
void attention_kernel(const float* __restrict__ q,
                      const float* __restrict__ vmat,
                      const float* __restrict__ kmat,
                      const float* __restrict__ scale_p,
                      float* __restrict__ attn_out,
                      float* __restrict__ upd_part,
                      float* __restrict__ asum_part)
{
  __shared__ float ks[V_][C_];
  const int b    = blockIdx.y;
  const int blk  = blockIdx.x;
  const int tid  = threadIdx.x;
  const int lane = tid & 31;
  const int wave = tid >> 5;

  for (int i = tid; i < V_ * C_; i += 256)
    ks[i >> 8][i & 255] = kmat[(size_t)b * V_ * C_ + i];
  __syncthreads();

  const float scale = scale_p[0];
  float accu[V_][8];
  float asum[V_];
  #pragma unroll
  for (int s = 0; s < V_; ++s) {
    asum[s] = 0.f;
    #pragma unroll
    for (int j = 0; j < 8; ++j) accu[s][j] = 0.f;
  }

  const int l0 = blk * TL_;
  for (int t = wave; t < TL_; t += 8) {
    const int l = l0 + t;
    const float* qrow = q + ((size_t)b * L_ + l) * C_;
    v4f q0 = ((const v4f*)qrow)[lane];        // cols lane*4 .. +3
    v4f q1 = ((const v4f*)qrow)[lane + 32];   // cols 128+lane*4 .. +3

    float an[V_];
    float mx = -1e30f;
    #pragma unroll
    for (int s = 0; s < V_; ++s) {
      const float* k0 = &ks[s][lane * 4];
      const float* k1 = &ks[s][128 + lane * 4];
      float d = q0.x * k0[0] + q0.y * k0[1] + q0.z * k0[2] + q0.w * k0[3]
              + q1.x * k1[0] + q1.y * k1[1] + q1.z * k1[2] + q1.w * k1[3];
      #pragma unroll
      for (int o = 16; o > 0; o >>= 1) d += __shfl_xor(d, o, 32);
      an[s] = d * scale;
      mx = fmaxf(mx, an[s]);
    }
    float se = 0.f;
    #pragma unroll
    for (int s = 0; s < V_; ++s) { an[s] = expf(an[s] - mx); se += an[s]; }
    const float rse = 1.f / se;
    #pragma unroll
    for (int s = 0; s < V_; ++s) an[s] *= rse;

    if (lane == 0) {
      #pragma unroll
      for (int s = 0; s < V_; ++s)
        attn_out[((size_t)b * L_ + l) * V_ + s] = an[s];
    }

    const float* vrow = vmat + ((size_t)b * L_ + l) * C_;
    v4f v0 = ((const v4f*)vrow)[lane];
    v4f v1 = ((const v4f*)vrow)[lane + 32];
    #pragma unroll
    for (int s = 0; s < V_; ++s) {
      accu[s][0] += an[s] * v0.x; accu[s][1] += an[s] * v0.y;
      accu[s][2] += an[s] * v0.z; accu[s][3] += an[s] * v0.w;
      accu[s][4] += an[s] * v1.x; accu[s][5] += an[s] * v1.y;
      accu[s][6] += an[s] * v1.z; accu[s][7] += an[s] * v1.w;
      asum[s]    += an[s];
    }
  }

  const int pidx = blk * 8 + wave;  // 0..127 per batch
  float* up = upd_part + ((size_t)b * 128 + pidx) * (V_ * C_);
  #pragma unroll
  for (int s = 0; s < V_; ++s) {
    #pragma unroll
    for (int j = 0; j < 4; ++j) up[s * C_ + lane * 4 + j]       = accu[s][j];
    #pragma unroll
    for (int j = 0; j < 4; ++j) up[s * C_ + 128 + lane * 4 + j] = accu[s][4 + j];
  }
  if (lane == 0) {
    #pragma unroll
    for (int s = 0; s < V_; ++s)
      asum_part[((size_t)b * 128 + pidx) * V_ + s] = asum[s];
  }
}

// ---------------------------------------------------------------------------
// Kernel 5: reduce partials, normalize, LN, MLP, slots += ...
// ---------------------------------------------------------------------------
__global__ __launch_bounds__(256)
void update_kernel(const float* __restrict__ upd_part,
                   const float* __restrict__ asum_part,
                   const float* __restrict__ g_u,
                   const float* __restrict__ b_u,
                   const float* __restrict__ W1,
                   const float* __restrict__ b1,
                   const float* __restrict__ W2,
                   const float* __restrict__ b2,
                   float* __restrict__ slots,
                   float* __restrict__ slots_out)
{
  __shared__ float h[V_][C_];
  __shared__ float mid[V_][2 * C_];
  __shared__ float red[16];
  __shared__ float asums[V_];
  const int b   = blockIdx.x;
  const int tid = threadIdx.x;

  if (tid < V_) {
    float a = 0.f;
    for (int p = 0; p < 128; ++p) a += asum_part[((size_t)b * 128 + p) * V_ + tid];
    asums[tid] = a + EPS_;
  }
  __syncthreads();

  for (int s = 0; s < V_; ++s) {
    float u = 0.f;
    for (int p = 0; p < 128; ++p)
      u += upd_part[((size_t)b * 128 + p) * (V_ * C_) + s * C_ + tid];
    u /= asums[s];
    float ss = u, s2 = u * u;
    block_reduce_2(ss, s2, red);
    float m   = ss * (1.f / C_);
    float inv = rsqrtf(s2 * (1.f / C_) - m * m + LN_EPS_);
    h[s][tid] = (u - m) * inv * g_u[tid] + b_u[tid];
  }
  __syncthreads();

  for (int s = 0; s < V_; ++s)
    for (int j = tid; j < 2 * C_; j += 256) {
      float a = b1[j];
      for (int c = 0; c < C_; ++c) a += h[s][c] * W1[(size_t)c * (2 * C_) + j];
      mid[s][j] = fmaxf(a, 0.f);
    }
  __syncthreads();

  for (int s = 0; s < V_; ++s) {
    float a = b2[tid];
    for (int j = 0; j < 2 * C_; ++j) a += mid[s][j] * W2[(size_t)j * C_ + tid];
    const size_t idx = ((size_t)b * V_ + s) * C_ + tid;
    const float nv = slots[idx] + a;
    slots[idx] = nv;
    if (slots_out) slots_out[idx] = nv;
  }
}

// ---------------------------------------------------------------------------
extern "C" void kernel_launch(void* const* d_in, const int* in_sizes, int n_in,
                              void* d_out, int out_size, void* d_ws, size_t ws_size,
                              hipStream_t stream)
{
  (void)in_sizes; (void)n_in; (void)out_size; (void)ws_size;
  const float* feat   = (const float*)d_in[0];
  const float* noise  = (const float*)d_in[1];
  const float* Wq     = (const float*)d_in[2];
  const float* Wk     = (const float*)d_in[3];
  const float* Wv     = (const float*)d_in[4];
  const float* scale  = (const float*)d_in[5];
  const float* g_in   = (const float*)d_in[6];
  const float* b_in   = (const float*)d_in[7];
  const float* g_sl   = (const float*)d_in[8];
  const float* b_sl   = (const float*)d_in[9];
  const float* g_up   = (const float*)d_in[10];
  const float* b_up   = (const float*)d_in[11];
  const float* W1     = (const float*)d_in[12];
  const float* b1     = (const float*)d_in[13];
  const float* W2     = (const float*)d_in[14];
  const float* b2     = (const float*)d_in[15];
  const float* nscale = (const float*)d_in[16];

  float* ws = (float*)d_ws;
  size_t off = 0;
  float* q         = ws + off; off += (size_t)B_ * L_ * C_;
  float* v         = ws + off; off += (size_t)B_ * L_ * C_;
  float* base_part = ws + off; off += (size_t)B_ * 64 * C_;
  float* slots     = ws + off; off += (size_t)B_ * V_ * C_;
  float* kbuf      = ws + off; off += (size_t)B_ * V_ * C_;
  float* upd_part  = ws + off; off += (size_t)B_ * 128 * V_ * C_;
  float* asum_part = ws + off; off += (size_t)B_ * 128 * V_;

  float* slots_out = (float*)d_out;
  float* attn_out  = (float*)d_out + (size_t)B_ * V_ * C_;

  proj_qv_kernel<<<dim3(L_ / 64, B_), 256, 0, stream>>>(
      feat, Wq, Wv, g_in, b_in, q, v, base_part);
  slots_init_kernel<<<B_, 256, 0, stream>>>(base_part, noise, nscale, slots);

  for (int t = 0; t < T_; ++t) {
    compute_k_kernel<<<B_, 256, 0, stream>>>(slots, Wk, g_sl, b_sl, kbuf);
    attention_kernel<<<dim3(L_ / TL_, B_), 256, 0, stream>>>(
        q, v, kbuf, scale, attn_out, upd_part, asum_part);
    update_kernel<<<B_, 256, 0, stream>>>(
        upd_part, asum_part, g_up, b_up, W1, b1, W2, b2,
        slots, (t == T_ - 1) ? slots_out : nullptr);
  }
}